// HOGLayer_32573031973414
// MI455X (gfx1250) — compile-verified
//
#include <hip/hip_runtime.h>
#include <math.h>
#include <stdint.h>

#define NBINS 10
#define HWI 512
#define CELLS 64          // cells per image dim (512/8)
#define NB 63             // block positions per dim (64-2+1)
#define NIMG 32
#define NBK 320           // n*bins = 32*10 reduction depth
#define NCELLTOT 4096     // 64*64

typedef __attribute__((ext_vector_type(2))) float v2f;
typedef __attribute__((ext_vector_type(8))) float v8f;

// ---------------------------------------------------------------------------
// CDNA5 async global->LDS copy (ASYNCcnt-tracked, bypasses VGPRs).
// Builtin is declared on this toolchain with (int AS1*, int AS3*, Imm, Imm).
// ---------------------------------------------------------------------------
typedef __attribute__((address_space(1))) int g_as1_int;
typedef __attribute__((address_space(3))) int l_as3_int;

__device__ __forceinline__ void async_load_f32_to_lds(const float* gptr,
                                                      float* lds_ptr) {
#if __has_builtin(__builtin_amdgcn_global_load_async_to_lds_b32)
  __builtin_amdgcn_global_load_async_to_lds_b32(
      (g_as1_int*)((float*)gptr),       // drop const, cast to AS1 int*
      (l_as3_int*)lds_ptr,              // generic(shared) -> AS3 int*
      /*offset=*/0, /*cpol=*/0);
#else
  uint32_t lds_off = (uint32_t)(uintptr_t)lds_ptr;  // addr[31:0] = LDS offset
  asm volatile("global_load_async_to_lds_b32 %0, %1, off"
               :: "v"(lds_off), "v"(gptr)
               : "memory");
#endif
}

__device__ __forceinline__ void wait_async_zero() {
#if __has_builtin(__builtin_amdgcn_s_wait_asynccnt)
  __builtin_amdgcn_s_wait_asynccnt(0);
#else
  asm volatile("s_wait_asynccnt 0" ::: "memory");
#endif
}

// ---------------------------------------------------------------------------
// Kernel 1: fused Sobel conv + soft histogram + 8x8 average pool.
// One 256-thread block handles a 32x32 pixel tile = 4x4 cells.
// Tile staging uses GLOBAL_LOAD_ASYNC_TO_LDS (memory -> LDS direct).
// ---------------------------------------------------------------------------
__global__ void hog_cells_kernel(const float* __restrict__ img,
                                 const float* __restrict__ w,
                                 float* __restrict__ cells) {
  __shared__ float tile[34 * 35];      // +1 col pad vs 64 banks
  __shared__ float hist[16 * NBINS];
  const int tid = threadIdx.x;
  const int n  = blockIdx.z;
  const int by = blockIdx.y * 32;
  const int bx = blockIdx.x * 32;

  if (tid < 16 * NBINS) hist[tid] = 0.0f;

  const float* imn = img + (size_t)n * HWI * HWI;
  for (int i = tid; i < 34 * 34; i += 256) {
    int r = i / 34, c = i % 34;
    int gy = by + r - 1, gx = bx + c - 1;
    float* dst = &tile[r * 35 + c];
    if (gy >= 0 && gy < HWI && gx >= 0 && gx < HWI) {
      async_load_f32_to_lds(&imn[gy * HWI + gx], dst);   // EXEC-masked per lane
    } else {
      *dst = 0.0f;                                       // zero halo (ds_store)
    }
  }

  // 3x3 cross-correlation weights (2 output channels), broadcast from L2
  float w0[9], w1[9];
#pragma unroll
  for (int k = 0; k < 9; ++k) { w0[k] = w[k]; w1[k] = w[9 + k]; }

  wait_async_zero();       // this wave's async LDS writes visible
  __syncthreads();         // all waves' tile writes visible

  const int px  = tid & 31;
  const int py0 = (tid >> 5) * 4;
  const float binScale = (float)NBINS / 3.14159265358979323846f;

  for (int q = 0; q < 4; ++q) {
    int py = py0 + q;
    int lr = py + 1, lc = px + 1;
    float g0 = 0.0f, g1 = 0.0f;
#pragma unroll
    for (int ky = 0; ky < 3; ++ky)
#pragma unroll
      for (int kx = 0; kx < 3; ++kx) {
        float p = tile[(lr - 1 + ky) * 35 + (lc - 1 + kx)];
        g0 = fmaf(w0[ky * 3 + kx], p, g0);
        g1 = fmaf(w1[ky * 3 + kx], p, g1);
      }
    float mag  = sqrtf(g0 * g0 + g1 * g1);
    float ph   = atan2f(g0, g1);          // phase = atan2(ch0, ch1)
    float pbin = ph * binScale;           // in (-10, 10]
    int fl = (int)floorf(pbin);
    int ce = (int)ceilf(pbin);
    int flb = fl % NBINS; if (flb < 0) flb += NBINS;   // python-style mod
    int ceb = ce % NBINS; if (ceb < 0) ceb += NBINS;
    int cell = ((py >> 3) << 2) + (px >> 3);
    if (flb == ceb) {
      atomicAdd(&hist[cell * NBINS + flb], 1.0f);      // mag + (1-mag)
    } else {
      atomicAdd(&hist[cell * NBINS + flb], mag);
      atomicAdd(&hist[cell * NBINS + ceb], 1.0f - mag);
    }
  }
  __syncthreads();

  for (int i = tid; i < 16 * NBINS; i += 256) {
    int cell = i / NBINS, b = i % NBINS;
    int r = blockIdx.y * 4 + (cell >> 2);
    int c = blockIdx.x * 4 + (cell & 3);
    cells[(((size_t)n * NBINS + b) * CELLS + r) * CELLS + c] = hist[i] * (1.0f / 64.0f);
  }
}

// ---------------------------------------------------------------------------
// Kernel 2: cellsq[r,c] = sum over (n,b) of cells^2, via V_WMMA_F32_16X16X4_F32.
// A = all-ones 16x4 => every row of D holds the K-reduction of B's columns,
// so the sum is invariant to the exact K-slot layout in B. One wave = 16 cells,
// 80 chained WMMA accumulates (K depth 320). EXEC all-1s before the WMMA loop.
// ---------------------------------------------------------------------------
__global__ void cellsq_wmma_kernel(const float* __restrict__ cells,
                                   float* __restrict__ cellsq) {
  const int lane = threadIdx.x;      // block = exactly one wave32
  const int col  = lane & 15;
  const int half = lane >> 4;
  const int cell = blockIdx.x * 16 + col;

  v8f acc = {0.f, 0.f, 0.f, 0.f, 0.f, 0.f, 0.f, 0.f};
  v2f a;  a.x = 1.0f; a.y = 1.0f;    // ones A-matrix: D[m,n] = sum_k B[k,n]

  for (int t = 0; t < NBK / 4; ++t) {
    int k0 = t * 4 + half * 2;
    float x0 = cells[(size_t)k0 * NCELLTOT + cell];
    float x1 = cells[(size_t)(k0 + 1) * NCELLTOT + cell];
    v2f b;  b.x = x0 * x0;  b.y = x1 * x1;
    acc = __builtin_amdgcn_wmma_f32_16x16x4_f32(
        /*neg_a=*/false, a, /*neg_b=*/false, b,
        /*c_mod=*/(short)0, acc, /*reuse_a=*/false, /*reuse_b=*/false);
  }
  // D VGPR0: lanes 0-15 hold (M=0, N=lane) -> column sums
  if (lane < 16) cellsq[cell] = acc[0];
}

// ---------------------------------------------------------------------------
// Kernel 3: s[br,bc] = 2x2 sum of cellsq; invS = rsqrt(s + eps^2)
// ---------------------------------------------------------------------------
__global__ void blocknorm1_kernel(const float* __restrict__ cellsq,
                                  float* __restrict__ invS) {
  int i = blockIdx.x * blockDim.x + threadIdx.x;
  if (i >= NB * NB) return;
  int br = i / NB, bc = i % NB;
  float s = cellsq[br * CELLS + bc]       + cellsq[br * CELLS + bc + 1] +
            cellsq[(br + 1) * CELLS + bc] + cellsq[(br + 1) * CELLS + bc + 1];
  invS[i] = rsqrtf(s + 1e-10f);           // EPS^2 = (1e-5)^2
}

// ---------------------------------------------------------------------------
// Kernel 4: s2[br,bc] = sum over (n,b,2x2) of min(cell*invS, 0.2)^2
// ---------------------------------------------------------------------------
__global__ void blocknorm2_kernel(const float* __restrict__ cells,
                                  const float* __restrict__ invS,
                                  float* __restrict__ invS2) {
  __shared__ float red[256];
  const int p  = blockIdx.x;             // 0..3968
  const int br = p / NB, bc = p % NB;
  const float is = invS[p];
  float sum = 0.0f;
  for (int i = threadIdx.x; i < NBK * 4; i += 256) {
    int nb = i >> 2;
    int q  = i & 3;
    int r = br + (q >> 1), c = bc + (q & 1);
    float v = cells[(size_t)nb * NCELLTOT + r * CELLS + c] * is;
    v = fminf(v, 0.2f);
    sum += v * v;
  }
  red[threadIdx.x] = sum;
  __syncthreads();
  for (int s = 128; s > 0; s >>= 1) {
    if (threadIdx.x < s) red[threadIdx.x] += red[threadIdx.x + s];
    __syncthreads();
  }
  if (threadIdx.x == 0) invS2[p] = rsqrtf(red[0] + 1e-10f);
}

// ---------------------------------------------------------------------------
// Kernel 5: expand to the (n, bins, 2, 2, 63, 63) feature vector.
// ---------------------------------------------------------------------------
__global__ void hog_out_kernel(const float* __restrict__ cells,
                               const float* __restrict__ invS,
                               const float* __restrict__ invS2,
                               float* __restrict__ out, int total) {
  int idx = blockIdx.x * blockDim.x + threadIdx.x;
  if (idx >= total) return;
  int bc = idx % NB;
  int t  = idx / NB;
  int br = t % NB;  t /= NB;
  int dj = t & 1;
  int di = (t >> 1) & 1;
  int nb = t >> 2;                        // n*NBINS + b
  int p  = br * NB + bc;
  float v = cells[(size_t)nb * NCELLTOT + (br + di) * CELLS + (bc + dj)] * invS[p];
  v = fminf(v, 0.2f);
  out[idx] = v * invS2[p];
}

// ---------------------------------------------------------------------------
extern "C" void kernel_launch(void* const* d_in, const int* in_sizes, int n_in,
                              void* d_out, int out_size, void* d_ws, size_t ws_size,
                              hipStream_t stream) {
  const float* img = (const float*)d_in[0];   // (32,1,512,512) f32
  const float* w   = (const float*)d_in[1];   // (2,1,3,3) f32

  float* ws     = (float*)d_ws;
  float* cells  = ws;                              // 32*10*64*64 = 1,310,720 f
  float* cellsq = cells + (size_t)NIMG * NBINS * CELLS * CELLS;  // 4096 f
  float* invS   = cellsq + NCELLTOT;               // 3969 f
  float* invS2  = invS + NB * NB;                  // 3969 f

  hog_cells_kernel<<<dim3(16, 16, NIMG), 256, 0, stream>>>(img, w, cells);
  cellsq_wmma_kernel<<<NCELLTOT / 16, 32, 0, stream>>>(cells, cellsq);
  blocknorm1_kernel<<<(NB * NB + 255) / 256, 256, 0, stream>>>(cellsq, invS);
  blocknorm2_kernel<<<NB * NB, 256, 0, stream>>>(cells, invS, invS2);

  int total = NIMG * NBINS * 2 * 2 * NB * NB;      // 5,080,320
  hog_out_kernel<<<(total + 255) / 256, 256, 0, stream>>>(cells, invS, invS2,
                                                          (float*)d_out, total);
}